// paraunitaryFIR_21079699489095
// MI455X (gfx1250) — compile-verified
//
#include <hip/hip_runtime.h>

// Problem constants (match reference)
#define KSTAGES 8
#define BATCH   32
#define FREQS   4096
#define NDIM    16

#define LDS_STRIDE 17                       // padded row stride -> conflict-free
#define MAT_LDS    (NDIM * LDS_STRIDE)      // 272 floats per staged matrix
#define WAVES_PER_BLOCK 8
#define THREADS (WAVES_PER_BLOCK * 32)

typedef __attribute__((ext_vector_type(2))) float v2f;
typedef __attribute__((ext_vector_type(4))) float v4f;
typedef __attribute__((ext_vector_type(8))) float v8f;

__global__ __launch_bounds__(THREADS)
void parau_fir_wmma(const float* __restrict__ U, const float* __restrict__ z,
                    const int* __restrict__ m, float* __restrict__ out) {
  __shared__ float lds[KSTAGES * MAT_LDS];  // ~8.5 KB of the 320 KB/WGP

  const int b    = blockIdx.x / (FREQS / WAVES_PER_BLOCK);
  const int fblk = blockIdx.x % (FREQS / WAVES_PER_BLOCK);
  const int wave = threadIdx.x >> 5;
  const int lane = threadIdx.x & 31;
  const int f    = fblk * WAVES_PER_BLOCK + wave;
  const int row  = lane & 15;               // n / M index carried by this lane
  const int half = lane >> 4;               // 0: lanes 0-15, 1: lanes 16-31

  // Stage all K stage-matrices for this batch b into LDS (row-major, padded).
  for (int idx = threadIdx.x; idx < KSTAGES * NDIM * NDIM; idx += THREADS) {
    const int k = idx >> 8;
    const int r = (idx >> 4) & 15;
    const int c = idx & 15;
    lds[k * MAT_LDS + r * LDS_STRIDE + c] =
        U[(((size_t)k * BATCH + b) * NDIM + r) * NDIM + c];
  }
  __syncthreads();

  const float zf = z[f];
  // d[n] = z_f^(-m[n]); this lane owns n == row
  const float dval = __powf(zf, -(float)m[row]);

  // dpat[2j+v]: lanes0-15 = d[4j+v], lanes16-31 = d[4j+v+2]
  // (exactly the per-lane-half scale needed for B-operand VGPR v of chunk j)
  float dpat[8];
#pragma unroll
  for (int j = 0; j < 4; ++j) {
#pragma unroll
    for (int v = 0; v < 2; ++v) {
      const float s0 = __shfl(dval, 4 * j + v, 32);
      const float s1 = __shfl(dval, 4 * j + v + 2, 32);
      dpat[j * 2 + v] = half ? s1 : s0;
    }
  }

  // A operands for stages 1..7: A = U_k^T, element (M=row, K) = U_k[K, row].
  // A-operand layout (16x4 f32): chunk j, vgpr v holds K = 4j + v + 2*half.
  float a[KSTAGES - 1][4][2];
#pragma unroll
  for (int k = 1; k < KSTAGES; ++k)
#pragma unroll
    for (int j = 0; j < 4; ++j)
#pragma unroll
      for (int v = 0; v < 2; ++v)
        a[k - 1][j][v] =
            lds[k * MAT_LDS + (4 * j + v + 2 * half) * LDS_STRIDE + row];

  // T = (D_f U_0)^T in C/D layout: VGPR v holds T(v+8*half, row)
  //   = d[row] * U_0[row, v + 8*half]
  float T[8];
#pragma unroll
  for (int v = 0; v < 8; ++v)
    T[v] = dval * lds[row * LDS_STRIDE + v + 8 * half];

#pragma unroll
  for (int k = 1; k < KSTAGES; ++k) {
    // Relayout running product: B = D_f * T, B chunk j vgpr v wants
    // row K0=4j+v in lanes0-15 and row K1=4j+v+2 in lanes16-31.
    float Bv[8];
#pragma unroll
    for (int j = 0; j < 4; ++j) {
#pragma unroll
      for (int v = 0; v < 2; ++v) {
        const int K0 = 4 * j + v;
        const int K1 = 4 * j + v + 2;
        float val;
        if (K0 < 8) { // rows K0,K1 live in lanes0-15 of T[K0],T[K1]
          const float sw = __shfl_xor(T[K1], 16, 32); // -> lanes16-31
          val = half ? sw : T[K0];
        } else {      // rows live in lanes16-31 of T[K0-8],T[K1-8]
          const float sw = __shfl_xor(T[K0 - 8], 16, 32); // -> lanes0-15
          val = half ? T[K1 - 8] : sw;
        }
        Bv[j * 2 + v] = val * dpat[j * 2 + v];
      }
    }

    // T_new = U_k^T x (D_f T): 4 accumulating f32 WMMAs cover K=16.
    v8f acc = {};
#pragma unroll
    for (int j = 0; j < 4; ++j) {
      v2f av;
      av.x = a[k - 1][j][0];
      av.y = a[k - 1][j][1];
      v2f bv;
      bv.x = Bv[j * 2 + 0];
      bv.y = Bv[j * 2 + 1];
      acc = __builtin_amdgcn_wmma_f32_16x16x4_f32(
          /*neg_a=*/false, av, /*neg_b=*/false, bv,
          /*c_mod=*/(short)0, acc, /*reuse_a=*/false, /*reuse_b=*/false);
    }
#pragma unroll
    for (int v = 0; v < 8; ++v) T[v] = acc[v];
  }

  // T = P^T in C/D layout -> lane holds P(row, 8*half + v) for v=0..7:
  // 8 contiguous floats per lane -> two coalesced 128-bit stores.
  float* optr = out + (((size_t)b * FREQS + f) * NDIM + row) * NDIM + 8 * half;
  v4f lo = {T[0], T[1], T[2], T[3]};
  v4f hi = {T[4], T[5], T[6], T[7]};
  *(v4f*)(optr) = lo;
  *(v4f*)(optr + 4) = hi;
}

extern "C" void kernel_launch(void* const* d_in, const int* in_sizes, int n_in,
                              void* d_out, int out_size, void* d_ws, size_t ws_size,
                              hipStream_t stream) {
  (void)in_sizes; (void)n_in; (void)d_ws; (void)ws_size; (void)out_size;
  const float* U = (const float*)d_in[0];   // [K, B, N, N] f32
  const float* z = (const float*)d_in[1];   // [FREQ] f32
  const int*   m = (const int*)d_in[2];     // [N] i32
  float* out = (float*)d_out;               // [B, FREQ, N, N] f32

  const int blocks = BATCH * (FREQS / WAVES_PER_BLOCK); // 16384
  parau_fir_wmma<<<blocks, THREADS, 0, stream>>>(U, z, m, out);
}